// OnsetEmbedding_86285892976712
// MI455X (gfx1250) — compile-verified
//
#include <hip/hip_runtime.h>

typedef __attribute__((ext_vector_type(2))) float v2f;
typedef __attribute__((ext_vector_type(8))) float v8f;

#define D 128

// ---------------- zero workspace ----------------
__global__ void gnn_zero_kernel(float* __restrict__ p, long n) {
  long i = (long)blockIdx.x * blockDim.x + threadIdx.x;
  long stride = (long)gridDim.x * blockDim.x;
  for (; i < n; i += stride) p[i] = 0.0f;
}

// ---------------- edge scatter: one wave32 per edge ----------------
// lane L handles floats [4L, 4L+3] of the 128-wide row.
__global__ void gnn_edge_scatter_kernel(const float* __restrict__ x,
                                        const int* __restrict__ ei,
                                        float* __restrict__ acc,
                                        float* __restrict__ cnt,
                                        int E) {
  int gtid = blockIdx.x * blockDim.x + threadIdx.x;
  int e = gtid >> 5;
  int lane = threadIdx.x & 31;
  if (e >= E) return;
  int s = ei[e];        // edge_index[0][e]
  int d = ei[E + e];    // edge_index[1][e]
  const float4* xs = (const float4*)(x + (size_t)s * D);
  const float4* xd = (const float4*)(x + (size_t)d * D);
  float4 a = xs[lane];
  float4 b = xd[lane];
  float4 m;
  m.x = fabsf(a.x - b.x);
  m.y = fabsf(a.y - b.y);
  m.z = fabsf(a.z - b.z);
  m.w = fabsf(a.w - b.w);
  float* ap = acc + (size_t)s * D + lane * 4;
  unsafeAtomicAdd(ap + 0, m.x);
  unsafeAtomicAdd(ap + 1, m.y);
  unsafeAtomicAdd(ap + 2, m.z);
  unsafeAtomicAdd(ap + 3, m.w);
  if (lane == 0) unsafeAtomicAdd(cnt + s, 1.0f);
}

// ---------------- normalized GEMM via V_WMMA_F32_16X16X4_F32 ----------------
// out[i, j] = b[j] + sum_k ((x[i,k]+acc[i,k]) / (cnt[i]+1)) * W[j,k]
// Block = 256 threads = 8 waves; wave w handles rows [blk*128 + w*16, +16),
// all 128 output columns (8 tiles of 16), K swept in steps of 4.
//
// LDS layout: K-pair packed transpose of W so each B fragment (2 consecutive
// K values for one output column) is ONE aligned ds_load_b64:
//   Wp[(k>>1) * 128 + n] = { W[n][k], W[n][k+1] }      (k even)
__launch_bounds__(256)
__global__ void gnn_gemm_wmma_kernel(const float* __restrict__ x,
                                     const float* __restrict__ acc,
                                     const float* __restrict__ cnt,
                                     const float* __restrict__ W,
                                     const float* __restrict__ bias,
                                     float* __restrict__ out,
                                     int N) {
  __shared__ float2 Wp[(D / 2) * D];   // 64 x 128 float2 = 64 KB

  int tid  = threadIdx.x;
  int lane = tid & 31;
  int wv   = tid >> 5;
  int l16  = lane & 15;
  int hi   = lane >> 4;         // 0: lanes 0-15, 1: lanes 16-31

  // Stage W into K-pair-packed transposed LDS layout.
  for (int i = tid; i < D * (D / 4); i += 256) {
    int n  = i >> 5;            // row of W (output feature)
    int kc = (i & 31) << 2;     // col of W (input feature), multiple of 4
    float4 v = *(const float4*)(W + n * D + kc);
    Wp[((kc >> 1) + 0) * D + n] = make_float2(v.x, v.y);
    Wp[((kc >> 1) + 1) * D + n] = make_float2(v.z, v.w);
  }
  __syncthreads();

  int row0 = blockIdx.x * 128 + wv * 16;
  // A-fragment row for this lane (clamped so loads stay in-bounds; stores guarded).
  int g = row0 + l16;
  if (g > N - 1) g = N - 1;
  float invc = 1.0f / (cnt[g] + 1.0f);
  const float* xr = x   + (size_t)g * D;
  const float* ar = acc + (size_t)g * D;

  v8f c[8];
#pragma unroll
  for (int t = 0; t < 8; ++t) {
    float bv = bias[t * 16 + l16];      // bias depends only on column
#pragma unroll
    for (int r = 0; r < 8; ++r) c[t][r] = bv;
  }

  for (int kk = 0; kk < D; kk += 4) {
    // This lane-half covers K = {kk+2*hi, kk+2*hi+1} -> packed pair row:
    int kh = (kk >> 1) + hi;
    float2 xv = *(const float2*)(xr + kk + 2 * hi);
    float2 av = *(const float2*)(ar + kk + 2 * hi);
    v2f a;
    a.x = (xv.x + av.x) * invc;
    a.y = (xv.y + av.y) * invc;

    // Load all 8 B fragments first (8 x ds_load_b64), then burst 8 WMMAs.
    v2f bf[8];
#pragma unroll
    for (int t = 0; t < 8; ++t) {
      float2 p = Wp[kh * D + t * 16 + l16];
      bf[t].x = p.x;
      bf[t].y = p.y;
    }
#pragma unroll
    for (int t = 0; t < 8; ++t) {
      c[t] = __builtin_amdgcn_wmma_f32_16x16x4_f32(
          false, a, false, bf[t], (short)0, c[t], false, false);
    }
  }

#pragma unroll
  for (int t = 0; t < 8; ++t) {
#pragma unroll
    for (int r = 0; r < 8; ++r) {
      int grow = row0 + r + 8 * hi;     // C/D: VGPR r -> row r (low lanes) / r+8 (high)
      if (grow < N) out[(size_t)grow * D + t * 16 + l16] = c[t][r];
    }
  }
}

extern "C" void kernel_launch(void* const* d_in, const int* in_sizes, int n_in,
                              void* d_out, int out_size, void* d_ws, size_t ws_size,
                              hipStream_t stream) {
  const float* x    = (const float*)d_in[0];
  const int*   ei   = (const int*)d_in[1];
  const float* W    = (const float*)d_in[2];
  const float* bias = (const float*)d_in[3];
  float* out = (float*)d_out;

  int N = in_sizes[0] / D;      // 100000
  int E = in_sizes[1] / 2;      // 600000

  float* acc = (float*)d_ws;                 // [N, 128]
  float* cnt = acc + (size_t)N * D;          // [N]

  // 1) zero acc + cnt (N*129 floats)
  long nz = (long)N * (D + 1);
  int zb = 1024;
  int zg = (int)((nz + zb - 1) / zb);
  if (zg > 4096) zg = 4096;
  gnn_zero_kernel<<<zg, zb, 0, stream>>>(acc, nz);

  // 2) edge scatter: one wave32 per edge, 8 edges per 256-thread block
  int eb = 256;
  int eg = (E * 32 + eb - 1) / eb;
  gnn_edge_scatter_kernel<<<eg, eb, 0, stream>>>(x, ei, acc, cnt, E);

  // 3) normalize + GEMM + bias via f32 WMMA; 128 rows per block
  int gg = (N + 127) / 128;
  gnn_gemm_wmma_kernel<<<gg, 256, 0, stream>>>(x, acc, cnt, W, bias, out, N);
}